// SimpleTransferableMLCV_64201171141456
// MI455X (gfx1250) — compile-verified
//
#include <hip/hip_runtime.h>
#include <hip/hip_bf16.h>
#include <math.h>

typedef __attribute__((ext_vector_type(2))) float v2f;
typedef __attribute__((ext_vector_type(8))) float v8f;

#define NB    128   // batch
#define NN    100   // nodes
#define NP    4950  // pairs = N*(N-1)/2
#define DM    128   // d_model
#define LDSP  132   // padded LDS row stride (kills stride-128 bank conflicts)

// ---------------------------------------------------------------------------
// Kernel 1: pe_row[n][dd] = sum_j make_pe(N,d)[n,j,dd]   (fixed [100,128] table)
// ---------------------------------------------------------------------------
__global__ void mlcv_pe_row(float* __restrict__ pe) {
    int idx = blockIdx.x * blockDim.x + threadIdx.x;
    if (idx >= NN * DM) return;
    int n  = idx / DM;
    int dd = idx % DM;
    float k  = 4.0f * (float)(dd >> 2);
    int   ch = dd & 3;
    float val;
    if (ch == 0) {
        float div0 = powf(10000.0f, k / (float)DM);
        val = (float)NN * sinf((float)n / div0);
    } else if (ch == 1) {
        float div0 = powf(10000.0f, k / (float)DM);
        val = (float)NN * cosf((float)n / div0);
    } else if (ch == 2) {
        float div2 = powf(10000.0f, (k + 2.0f) / (float)DM);
        float s = 0.0f;
        for (int j = 0; j < NN; ++j) s += sinf((float)j / div2);
        val = s;
    } else {
        float div3 = powf(10000.0f, (k + 3.0f) / (float)DM);
        float s = 0.0f;
        for (int j = 0; j < NN; ++j) s += cosf(fabsf((float)(n - j)) / div3);
        val = s;
    }
    pe[idx] = val;
}

// ---------------------------------------------------------------------------
// Kernel 2: S[b*N+n] = sum of cad over the 99 edges incident to node n.
// Affine-embedding collapse: row[b,n,:] = S[b,n]*w_dist + 99*b_dist.
// triu edge index (i<j): p = i*N - i*(i+1)/2 + (j-i-1)
// ---------------------------------------------------------------------------
__global__ void mlcv_edge_sum(const float* __restrict__ cad, float* __restrict__ S) {
    int idx = blockIdx.x * blockDim.x + threadIdx.x;
    if (idx >= NB * NN) return;
    int b = idx / NN;
    int n = idx % NN;
    const float* c = cad + (size_t)b * NP;
    float s = 0.0f;
    for (int m = 0; m < NN; ++m) {
        if (m == n) continue;
        int i = m < n ? m : n;
        int j = m < n ? n : m;
        int p = i * NN - (i * (i + 1)) / 2 + (j - i - 1);
        s += c[p];
    }
    S[idx] = s;
}

// ---------------------------------------------------------------------------
// Kernel 3: fused x1d build + 2-layer MLP via V_WMMA_F32_16X16X4_F32.
// Workgroup = 256 threads = 8 wave32; owns a 16-row M-tile of the 12800x128
// activation matrix. Wave w computes N-columns [16w,16w+16). 32 K-steps of
// f32 WMMA per layer; h1 staged through LDS (padded).
// ---------------------------------------------------------------------------
__global__ void __launch_bounds__(256)
mlcv_gemm2(const float* __restrict__ S,   const float* __restrict__ pe,
           const float* __restrict__ wD,  const float* __restrict__ bD,
           const float* __restrict__ W1,  const float* __restrict__ B1,
           const float* __restrict__ W2,  const float* __restrict__ B2,
           float* __restrict__ h2out) {
    __shared__ float xs[16][LDSP];
    __shared__ float hs[16][LDSP];

    const int row0 = blockIdx.x * 16;         // 800 blocks * 16 rows = 12800
    const int tid  = threadIdx.x;

    // Build x1d tile in LDS: (S*w_dist + 99*b_dist + pe_row[n]) / N
    for (int idx = tid; idx < 16 * DM; idx += 256) {
        int m  = idx >> 7;
        int dd = idx & (DM - 1);
        int row = row0 + m;
        int n   = row % NN;
        xs[m][dd] = (S[row] * wD[dd] + 99.0f * bD[dd] + pe[n * DM + dd]) * (1.0f / (float)NN);
    }
    __syncthreads();

    const int lane = tid & 31;
    const int hi   = lane >> 4;               // 0: K={0,1}, 1: K={2,3}
    const int ln   = lane & 15;
    const int nc   = (tid >> 5) * 16;         // this wave's N-column base

    // ---- layer 1: h1 = relu(x1d @ W1 + B1) ----
    v8f c = {};
    for (int k0 = 0; k0 < DM; k0 += 4) {
        v2f a, b;
        a.x = xs[ln][k0 + 2 * hi];            // A[M=ln][K]
        a.y = xs[ln][k0 + 2 * hi + 1];
        b.x = W1[(k0 + 2 * hi) * DM + nc + ln];       // B[K][N=nc+ln]
        b.y = W1[(k0 + 2 * hi + 1) * DM + nc + ln];
        c = __builtin_amdgcn_wmma_f32_16x16x4_f32(false, a, false, b,
                                                  (short)0, c, false, false);
    }
    {
        float bv = B1[nc + ln];
        #pragma unroll
        for (int r = 0; r < 8; ++r) {         // C: VGPR r -> M=r+8*hi, N=ln
            float v = c[r] + bv;
            hs[r + 8 * hi][nc + ln] = v > 0.0f ? v : 0.0f;
        }
    }
    __syncthreads();

    // ---- layer 2: h2 = relu(h1 @ W2 + B2) ----
    v8f c2 = {};
    for (int k0 = 0; k0 < DM; k0 += 4) {
        v2f a, b;
        a.x = hs[ln][k0 + 2 * hi];
        a.y = hs[ln][k0 + 2 * hi + 1];
        b.x = W2[(k0 + 2 * hi) * DM + nc + ln];
        b.y = W2[(k0 + 2 * hi + 1) * DM + nc + ln];
        c2 = __builtin_amdgcn_wmma_f32_16x16x4_f32(false, a, false, b,
                                                   (short)0, c2, false, false);
    }
    {
        float bv = B2[nc + ln];
        #pragma unroll
        for (int r = 0; r < 8; ++r) {
            float v = c2[r] + bv;
            h2out[(size_t)(row0 + r + 8 * hi) * DM + nc + ln] = v > 0.0f ? v : 0.0f;
        }
    }
}

// ---------------------------------------------------------------------------
// Kernel 4: mean-pool over n (fixed order -> deterministic) + 128->64->2 head.
// One block per batch element, 128 threads.
// ---------------------------------------------------------------------------
__global__ void __launch_bounds__(128)
mlcv_pool_head(const float* __restrict__ h2,
               const float* __restrict__ Wo1, const float* __restrict__ bo1,
               const float* __restrict__ Wo2, const float* __restrict__ bo2,
               float* __restrict__ out) {
    __shared__ float p[DM];
    __shared__ float q[64];
    int b = blockIdx.x;
    int t = threadIdx.x;

    const float* base = h2 + (size_t)b * NN * DM;
    float acc = 0.0f;
    for (int n = 0; n < NN; ++n) acc += base[n * DM + t];   // coalesced over t
    p[t] = acc * (1.0f / (float)NN);
    __syncthreads();

    if (t < 64) {
        float a = bo1[t];
        for (int d0 = 0; d0 < DM; ++d0) a += p[d0] * Wo1[d0 * 64 + t];
        q[t] = a > 0.0f ? a : 0.0f;
    }
    __syncthreads();

    if (t < 2) {
        float a = bo2[t];
        for (int j = 0; j < 64; ++j) a += q[j] * Wo2[j * 2 + t];
        out[b * 2 + t] = a;
    }
}

// ---------------------------------------------------------------------------
extern "C" void kernel_launch(void* const* d_in, const int* in_sizes, int n_in,
                              void* d_out, int out_size, void* d_ws, size_t ws_size,
                              hipStream_t stream) {
    const float* cad = (const float*)d_in[0];
    const float* wD  = (const float*)d_in[1];
    const float* bD  = (const float*)d_in[2];
    const float* W1  = (const float*)d_in[3];
    const float* B1  = (const float*)d_in[4];
    const float* W2  = (const float*)d_in[5];
    const float* B2  = (const float*)d_in[6];
    const float* Wo1 = (const float*)d_in[7];
    const float* bo1 = (const float*)d_in[8];
    const float* Wo2 = (const float*)d_in[9];
    const float* bo2 = (const float*)d_in[10];
    float* out = (float*)d_out;

    float* ws = (float*)d_ws;
    float* pe = ws;                  // 100*128  = 12800 floats (reserve 16384)
    float* S  = ws + 16384;          // 128*100  = 12800 floats (reserve 16384)
    float* h2 = ws + 32768;          // 12800*128 = 1638400 floats (~6.25 MB)

    mlcv_pe_row  <<<(NN * DM + 255) / 256, 256, 0, stream>>>(pe);
    mlcv_edge_sum<<<(NB * NN + 255) / 256, 256, 0, stream>>>(cad, S);
    mlcv_gemm2   <<<(NB * NN) / 16, 256, 0, stream>>>(S, pe, wD, bD, W1, B1, W2, B2, h2);
    mlcv_pool_head<<<NB, 128, 0, stream>>>(h2, Wo1, bo1, Wo2, bo2, out);
}